// GTLayer_41283225649459
// MI455X (gfx1250) — compile-verified
//
#include <hip/hip_runtime.h>
#include <hip/hip_bf16.h>

typedef __attribute__((ext_vector_type(2))) float v2f;
typedef __attribute__((ext_vector_type(8))) float v8f;

#define LATDIM 128
#define HEADS 8
#define HDIM 16

__global__ __launch_bounds__(256) void zero_kernel(float* __restrict__ p, int n) {
    int i = blockIdx.x * blockDim.x + threadIdx.x;
    if (i < n) p[i] = 0.0f;
}

// One wave per (16-row tile, matrix) pair. Accumulates all 8 column tiles of
// the 128-wide output with V_WMMA_F32_16X16X4_F32 (fp32-exact matrix op).
__global__ __launch_bounds__(256) void qkv_gemm_wmma(
    const float* __restrict__ embeds,
    const float* __restrict__ Wq, const float* __restrict__ Wk,
    const float* __restrict__ Wv,
    float* __restrict__ Q, float* __restrict__ K, float* __restrict__ V,
    int rowTiles, int n_nodes)
{
    const int lane = threadIdx.x & 31;
    const int wave = threadIdx.x >> 5;
    const int gw   = blockIdx.x * (blockDim.x >> 5) + wave;
    const int mat  = gw % 3;
    const int rt   = gw / 3;
    if (rt >= rowTiles) return;   // wave-uniform: EXEC stays all-1s for WMMA

    const float* __restrict__ W   = (mat == 0) ? Wq : (mat == 1) ? Wk : Wv;
    float*       __restrict__ Out = (mat == 0) ? Q  : (mat == 1) ? K  : V;

    const int m     = lane & 15;      // M (for A) / N (for B,C)
    const int hi    = lane >> 4;      // half-wave select
    const int khalf = hi << 1;        // K sub-offset: 0 or 2

    // A row for this lane (clamp keeps loads in-bounds without touching EXEC)
    int arowIdx = rt * 16 + m;
    if (arowIdx >= n_nodes) arowIdx = n_nodes - 1;
    const float* __restrict__ arow = embeds + (size_t)arowIdx * LATDIM;

    v8f acc[8] = {};

#pragma unroll 2
    for (int k0 = 0; k0 < LATDIM; k0 += 4) {
        // A fragment: 16x4 f32 -> 2 VGPRs
        v2f a;
        a.x = arow[k0 + khalf];
        a.y = arow[k0 + khalf + 1];
        const float* __restrict__ w0 = W + (size_t)(k0 + khalf) * LATDIM + m;
#pragma unroll
        for (int t = 0; t < 8; ++t) {
            // B fragment: 4x16 f32 -> 2 VGPRs (row k striped across lanes)
            v2f b;
            b.x = w0[t * 16];
            b.y = w0[t * 16 + LATDIM];
            acc[t] = __builtin_amdgcn_wmma_f32_16x16x4_f32(
                false, a, false, b, (short)0, acc[t], false, false);
        }
    }

    // C/D layout: VGPR r -> M = r + 8*hi, N = lane&15
    const int orow = rt * 16 + hi * 8;
#pragma unroll
    for (int t = 0; t < 8; ++t) {
#pragma unroll
        for (int r = 0; r < 8; ++r) {
            int row = orow + r;
            if (row < n_nodes)
                Out[(size_t)row * LATDIM + t * 16 + m] = acc[t][r];
        }
    }
}

// One thread per (edge, head): 16-dim dot product, clip, exp, softmax-denom
// scatter-add.
__global__ __launch_bounds__(256) void edge_scores(
    const float* __restrict__ Q, const float* __restrict__ K,
    const int* __restrict__ rows, const int* __restrict__ cols,
    float* __restrict__ expAtt, float* __restrict__ attNorm, int n_edges)
{
    int idx = blockIdx.x * blockDim.x + threadIdx.x;
    if (idx >= n_edges * HEADS) return;
    int e = idx >> 3;
    int h = idx & 7;
    int r = rows[e], c = cols[e];
    const float4* qp = (const float4*)(Q + (size_t)r * LATDIM + h * HDIM);
    const float4* kp = (const float4*)(K + (size_t)c * LATDIM + h * HDIM);
    float att = 0.0f;
#pragma unroll
    for (int j = 0; j < 4; ++j) {
        float4 q4 = qp[j];
        float4 k4 = kp[j];
        att += q4.x * k4.x + q4.y * k4.y + q4.z * k4.z + q4.w * k4.w;
    }
    att = fminf(10.0f, fmaxf(-10.0f, att));
    float ea = __expf(att);
    expAtt[idx] = ea;
    unsafeAtomicAdd(attNorm + (size_t)r * HEADS + h, ea);
}

// 32 lanes per edge; each lane owns 4 output floats (one quarter-head).
__global__ __launch_bounds__(256) void edge_aggregate(
    const float* __restrict__ V,
    const int* __restrict__ rows, const int* __restrict__ cols,
    const float* __restrict__ expAtt, const float* __restrict__ attNorm,
    float* __restrict__ out, int n_edges)
{
    int t = blockIdx.x * blockDim.x + threadIdx.x;
    int e = t >> 5;
    if (e >= n_edges) return;
    int lane = t & 31;
    int r = rows[e], c = cols[e];
    int h = lane >> 2;
    float a = expAtt[(size_t)e * HEADS + h] /
              (attNorm[(size_t)r * HEADS + h] + 1e-8f);
    float4 v4 = *(const float4*)(V + (size_t)c * LATDIM + lane * 4);
    float* op = out + (size_t)r * LATDIM + lane * 4;
    unsafeAtomicAdd(op + 0, a * v4.x);
    unsafeAtomicAdd(op + 1, a * v4.y);
    unsafeAtomicAdd(op + 2, a * v4.z);
    unsafeAtomicAdd(op + 3, a * v4.w);
}

extern "C" void kernel_launch(void* const* d_in, const int* in_sizes, int n_in,
                              void* d_out, int out_size, void* d_ws, size_t ws_size,
                              hipStream_t stream) {
    const float* embeds = (const float*)d_in[0];
    const float* Wq     = (const float*)d_in[1];
    const float* Wk     = (const float*)d_in[2];
    const float* Wv     = (const float*)d_in[3];
    const int*   rows   = (const int*)d_in[4];
    const int*   cols   = (const int*)d_in[5];

    const int N = in_sizes[0] / LATDIM;
    const int E = in_sizes[4];

    float* out = (float*)d_out;

    // Workspace layout (floats): Q | K | V | expAtt | attNorm
    float* ws      = (float*)d_ws;
    float* Qm      = ws;
    float* Km      = Qm + (size_t)N * LATDIM;
    float* Vm      = Km + (size_t)N * LATDIM;
    float* expAtt  = Vm + (size_t)N * LATDIM;
    float* attNorm = expAtt + (size_t)E * HEADS;

    // Zero accumulation targets (harness poisons them with 0xAA)
    {
        int n = N * LATDIM;
        zero_kernel<<<(n + 255) / 256, 256, 0, stream>>>(out, n);
        n = N * HEADS;
        zero_kernel<<<(n + 255) / 256, 256, 0, stream>>>(attNorm, n);
    }

    // Phase 1: Q/K/V node projections via fp32 WMMA
    const int rowTiles = (N + 15) / 16;
    const int nwaves   = rowTiles * 3;
    qkv_gemm_wmma<<<(nwaves + 7) / 8, 256, 0, stream>>>(
        embeds, Wq, Wk, Wv, Qm, Km, Vm, rowTiles, N);

    // Phase 2: per-(edge,head) scores + softmax denominators
    {
        int npairs = E * HEADS;
        edge_scores<<<(npairs + 255) / 256, 256, 0, stream>>>(
            Qm, Km, rows, cols, expAtt, attNorm, E);
    }

    // Phase 3: normalized weighted scatter-add of V into out
    {
        long nthreads = (long)E * 32;
        edge_aggregate<<<(int)((nthreads + 255) / 256), 256, 0, stream>>>(
            Vm, rows, cols, expAtt, attNorm, out, E);
    }
}